// MultiHeadEMA_38010460570378
// MI455X (gfx1250) — compile-verified
//
#include <hip/hip_runtime.h>
#include <math.h>

// MultiHeadEMA on MI455X (gfx1250, wave32).
//
// kernel k[d,l] = sum_n w_n q_n^l  =>  N=16-state EMA recurrence.
// Chunked state-space form (chunk=16, superblock=16 chunks=256 steps):
//   X[i,c] = x[t0 + 16c + i]
//   g      = G @ X         (G[n,j] = q_n^{15-j})
//   P      = per-row scan of g over chunks (factor q_n^16) -> S_prev per chunk
//   Y      = T @ X + M @ S_prev  (T = Toeplitz(k), M[i,n] = w_n q_n^{i+1})
// fp32 16x16x16 matmuls = 4 chained V_WMMA_F32_16X16X4_F32 each.
//
// Data movement: per-channel time series are 4B elements strided by B*D*4=16KB.
// TDM (tensor_load_to_lds / tensor_store_from_lds) performs the strided
// gather/scatter (2D descriptor, tile 1x256, stride 4096 elems); the 192MB L2
// absorbs the whole 64MB x so HBM sees each line once. TDM LDS padding
// (2 dwords per 16) makes the staged window conflict-free for ds_load_b64.

#define LL   4096
#define BB   4
#define DD   1024
#define NN   16
#define SBL  256          // superblock length
#define NSB  (LL / SBL)   // 16
#define NCH  (BB * DD)    // 4096 channels

// workspace layout (floats)
#define CSTR   1024
#define C_Q    0
#define C_W    16
#define C_Q16  32
#define C_L2   48
#define C_T    64
#define C_M    320
#define C_G    576
#define CONST_FLOATS ((size_t)DD * CSTR)
#define F_OFF   CONST_FLOATS
#define SIN_OFF (F_OFF + (size_t)NCH * NSB * NN)

// LDS staging: per-wave padded load window (256 data + 2 pad per 16) = 288 dw
#define LDSW      288
#define NWAVES    8
#define STORE_DW  (LDSW * NWAVES)          // store region starts here (dwords)
#define LDS_LOAD_BYTES  (LDSW * NWAVES * 4)              // 9216
#define LDS_ALL_BYTES   (LDS_LOAD_BYTES + NWAVES * SBL * 4)  // 17408

typedef float v2f __attribute__((ext_vector_type(2)));
typedef float v8f __attribute__((ext_vector_type(8)));
typedef unsigned int v4u __attribute__((ext_vector_type(4)));
typedef int v4i __attribute__((ext_vector_type(4)));
typedef int v8i __attribute__((ext_vector_type(8)));

static __device__ __forceinline__ v8f wmma4(v2f a, v2f b, v8f c) {
  return __builtin_amdgcn_wmma_f32_16x16x4_f32(false, a, false, b, (short)0, c,
                                               false, false);
}

// TDM 2D descriptor: data_size=4B, tile 1 x 256, tensor dims 1 x 256,
// dim0 stride 4096 elements (= B*D). Group1 word0:
//   load : data_size=2 | pad_en | pad_interval=3 (16dw) | pad_amount=1 (2dw)
//   store: data_size=2 (no padding on stores)
static __device__ __forceinline__ void tdm_load_chan(const float* gptr,
                                                     unsigned lds_byte_off) {
  const unsigned long long ga = (unsigned long long)(uintptr_t)gptr;
  v4u g0 = {1u, lds_byte_off, (unsigned)ga,
            (unsigned)(ga >> 32) | 0x80000000u};      // type=2 in bits 127:126
  v8i g1 = {(int)0x02D20000, 0x00010000, 0x01000000, 0x00010000,
            0x00000100, 4096, 0, 0};
  v4i z4 = {0, 0, 0, 0};
  v8i z8 = {0, 0, 0, 0, 0, 0, 0, 0};
  __builtin_amdgcn_tensor_load_to_lds(g0, g1, z4, z4, z8, 0);
}

static __device__ __forceinline__ void tdm_store_chan(float* gptr,
                                                      unsigned lds_byte_off) {
  const unsigned long long ga = (unsigned long long)(uintptr_t)gptr;
  v4u g0 = {1u, lds_byte_off, (unsigned)ga,
            (unsigned)(ga >> 32) | 0x80000000u};
  v8i g1 = {0x00020000, 0x00010000, 0x01000000, 0x00010000,
            0x00000100, 4096, 0, 0};
  v4i z4 = {0, 0, 0, 0};
  v8i z8 = {0, 0, 0, 0, 0, 0, 0, 0};
  __builtin_amdgcn_tensor_store_from_lds(g0, g1, z4, z4, z8, 0);
}

// ---------------------------------------------------------------- pass 0: constants
__global__ void k_const(const float* __restrict__ delta, const float* __restrict__ alpha,
                        const float* __restrict__ beta,  const float* __restrict__ gamma,
                        float* __restrict__ ws) {
  const int d = blockIdx.x;
  const int l = threadIdx.x;            // one wave per d
  __shared__ float kpow[16][17];
  __shared__ float warr[16];
  __shared__ float ktoe[16];
  float* cb = ws + (size_t)d * CSTR;

  if (l < 16) {
    const int n = l;
    const float dl = delta[d * NN + n];
    const float al = alpha[d * NN + n];
    const float be = beta [d * NN + n];
    const float ga = gamma[d * NN + n];
    const float p  = 1.0f / (1.0f + expf(-dl));
    const float sa = 1.0f / (1.0f + expf(-al));
    const float q  = 1.0f - p * sa;               // in (0,1)
    const float w  = p * be * ga * 0.25f;         // scale = 1/sqrt(16)
    float q2 = q * q, q4 = q2 * q2, q8 = q4 * q4, q16 = q8 * q8;
    cb[C_Q   + n] = q;
    cb[C_W   + n] = w;
    cb[C_Q16 + n] = q16;
    cb[C_L2  + n] = 16.0f * log2f(q);
    warr[n] = w;
    float pw = 1.0f;
    for (int t = 0; t <= 16; ++t) { kpow[n][t] = pw; pw *= q; }
  }
  __syncthreads();
  if (l < 16) {
    float s = 0.0f;
    #pragma unroll
    for (int n = 0; n < 16; ++n) s += warr[n] * kpow[n][l];
    ktoe[l] = s;
  }
  __syncthreads();
  const int m = l & 15, h = l >> 4;
  #pragma unroll
  for (int kb = 0; kb < 4; ++kb)
    #pragma unroll
    for (int v = 0; v < 2; ++v) {
      const int K   = 4 * kb + v + 2 * h;         // A-operand: K = v + 2*half
      const int idx = (kb * 2 + v) * 32 + l;
      cb[C_T + idx] = (m >= K) ? ktoe[m - K] : 0.0f;
      cb[C_M + idx] = warr[K] * kpow[K][m + 1];
      cb[C_G + idx] = kpow[m][15 - K];
    }
}

// ---------------------------------------------------------------- pass 1: local F
__global__ void k_pass1(const float* __restrict__ x, float* __restrict__ ws) {
  extern __shared__ float lds[];
  const int ch   = blockIdx.x;
  const int wave = threadIdx.x >> 5;
  const int sb   = blockIdx.y * NWAVES + wave;
  const int l    = threadIdx.x & 31;
  const int c    = l & 15, h = l >> 4;
  const int d    = ch & (DD - 1);
  const int b    = ch >> 10;
  const float* cb = ws + (size_t)d * CSTR;
  const size_t baseChan = (size_t)b * DD + d;
  const int t0 = sb * SBL;

  // async strided gather of this wave's 256-step window into padded LDS
  tdm_load_chan(x + (((size_t)t0) << 12) + baseChan, (unsigned)(wave * LDSW * 4));
  __builtin_amdgcn_s_wait_tensorcnt(0);
  asm volatile("" ::: "memory");

  const float* wb = lds + wave * LDSW;  // row stride 18 dwords (16 data + 2 pad)
  v2f Xb[4];
  #pragma unroll
  for (int kb = 0; kb < 4; ++kb) {
    const float2 p = *(const float2*)(wb + 18 * c + 4 * kb + 2 * h);
    Xb[kb].x = p.x; Xb[kb].y = p.y;
  }
  // g = G @ X
  v8f g = {0.f, 0.f, 0.f, 0.f, 0.f, 0.f, 0.f, 0.f};
  #pragma unroll
  for (int kb = 0; kb < 4; ++kb) {
    v2f a;
    a.x = cb[C_G + (kb * 2 + 0) * 32 + l];
    a.y = cb[C_G + (kb * 2 + 1) * 32 + l];
    g = wmma4(a, Xb[kb], g);
  }
  // inclusive scan over chunks c (rows n = j + 8h, factor q_n^16)
  float P[8], qs[8];
  #pragma unroll
  for (int j = 0; j < 8; ++j) { P[j] = g[j]; qs[j] = cb[C_Q16 + j + 8 * h]; }
  #pragma unroll
  for (int s = 1; s < 16; s <<= 1) {
    #pragma unroll
    for (int j = 0; j < 8; ++j) {
      const float up = __shfl_up(P[j], s, 16);
      P[j] += (c >= s) ? qs[j] * up : 0.0f;
      qs[j] *= qs[j];
    }
  }
  // F[n] = P[c=15]
  float* F = ws + F_OFF + (size_t)ch * (NSB * NN) + (size_t)sb * NN;
  if (c == 15) {
    #pragma unroll
    for (int j = 0; j < 8; ++j) F[j + 8 * h] = P[j];
  }
}

// ---------------------------------------------------------------- pass 2: global scan
__global__ void k_pass2(float* __restrict__ ws) {
  const int tid = blockIdx.x * blockDim.x + threadIdx.x;
  const int ch = tid >> 4, n = tid & 15;
  const int d  = ch & (DD - 1);
  const float q16 = ws[(size_t)d * CSTR + C_Q16 + n];
  float q256 = q16 * q16; q256 *= q256; q256 *= q256; q256 *= q256;
  const float* F = ws + F_OFF   + (size_t)ch * (NSB * NN);
  float*       S = ws + SIN_OFF + (size_t)ch * (NSB * NN);
  float s = 0.0f;
  S[n] = 0.0f;
  for (int sb = 1; sb < NSB; ++sb) {
    s = q256 * s + F[(sb - 1) * NN + n];
    S[sb * NN + n] = s;
  }
}

// ---------------------------------------------------------------- pass 3: outputs
__global__ void k_pass3(const float* __restrict__ x, const float* __restrict__ omega,
                        const float* __restrict__ ws, float* __restrict__ out) {
  extern __shared__ float lds[];
  const int ch   = blockIdx.x;
  const int wave = threadIdx.x >> 5;
  const int sb   = blockIdx.y * NWAVES + wave;
  const int l    = threadIdx.x & 31;
  const int c    = l & 15, h = l >> 4;
  const int d    = ch & (DD - 1);
  const int b    = ch >> 10;
  const float* cb = ws + (size_t)d * CSTR;
  const size_t baseChan = (size_t)b * DD + d;
  const int t0 = sb * SBL;

  tdm_load_chan(x + (((size_t)t0) << 12) + baseChan, (unsigned)(wave * LDSW * 4));
  __builtin_amdgcn_s_wait_tensorcnt(0);
  asm volatile("" ::: "memory");

  const float* wb = lds + wave * LDSW;
  // X in B-operand layout + C/D-layout copy (residual path)
  v2f Xb[4];
  #pragma unroll
  for (int kb = 0; kb < 4; ++kb) {
    const float2 p = *(const float2*)(wb + 18 * c + 4 * kb + 2 * h);
    Xb[kb].x = p.x; Xb[kb].y = p.y;
  }
  float Xd[8];
  #pragma unroll
  for (int jj = 0; jj < 4; ++jj) {
    const float2 p = *(const float2*)(wb + 18 * c + 8 * h + 2 * jj);
    Xd[2 * jj] = p.x; Xd[2 * jj + 1] = p.y;
  }

  // g = G @ X, scan -> P
  v8f g = {0.f, 0.f, 0.f, 0.f, 0.f, 0.f, 0.f, 0.f};
  #pragma unroll
  for (int kb = 0; kb < 4; ++kb) {
    v2f a;
    a.x = cb[C_G + (kb * 2 + 0) * 32 + l];
    a.y = cb[C_G + (kb * 2 + 1) * 32 + l];
    g = wmma4(a, Xb[kb], g);
  }
  float P[8], qs[8];
  #pragma unroll
  for (int j = 0; j < 8; ++j) { P[j] = g[j]; qs[j] = cb[C_Q16 + j + 8 * h]; }
  #pragma unroll
  for (int s = 1; s < 16; s <<= 1) {
    #pragma unroll
    for (int j = 0; j < 8; ++j) {
      const float up = __shfl_up(P[j], s, 16);
      P[j] += (c >= s) ? qs[j] * up : 0.0f;
      qs[j] *= qs[j];
    }
  }
  // S_prev[n,c] = q16^c * S_in[n] + P[c-1]
  const float* Sin = ws + SIN_OFF + (size_t)ch * (NSB * NN) + (size_t)sb * NN;
  float Sp[8];
  #pragma unroll
  for (int j = 0; j < 8; ++j) {
    float pm1 = __shfl_up(P[j], 1, 16);
    pm1 = (c >= 1) ? pm1 : 0.0f;
    const float qc = exp2f((float)c * cb[C_L2 + j + 8 * h]);
    Sp[j] = qc * Sin[j + 8 * h] + pm1;
  }
  // C/D layout -> B-operand tiles: half-swap + select
  float Su[8];
  #pragma unroll
  for (int j = 0; j < 8; ++j) Su[j] = __shfl_xor(Sp[j], 16, 32);
  const bool hi = (h == 1);
  v2f Bc[4];
  Bc[0].x = hi ? Su[2] : Sp[0];  Bc[0].y = hi ? Su[3] : Sp[1];
  Bc[1].x = hi ? Su[6] : Sp[4];  Bc[1].y = hi ? Su[7] : Sp[5];
  Bc[2].x = hi ? Sp[2] : Su[0];  Bc[2].y = hi ? Sp[3] : Su[1];
  Bc[3].x = hi ? Sp[6] : Su[4];  Bc[3].y = hi ? Sp[7] : Su[5];

  // Y = T @ X + M @ S_prev
  v8f acc = {0.f, 0.f, 0.f, 0.f, 0.f, 0.f, 0.f, 0.f};
  #pragma unroll
  for (int kb = 0; kb < 4; ++kb) {
    v2f a;
    a.x = cb[C_T + (kb * 2 + 0) * 32 + l];
    a.y = cb[C_T + (kb * 2 + 1) * 32 + l];
    acc = wmma4(a, Xb[kb], acc);
  }
  #pragma unroll
  for (int kb = 0; kb < 4; ++kb) {
    v2f a;
    a.x = cb[C_M + (kb * 2 + 0) * 32 + l];
    a.y = cb[C_M + (kb * 2 + 1) * 32 + l];
    acc = wmma4(a, Bc[kb], acc);
  }

  // silu(Y + omega*x) -> contiguous (in t) LDS window, then TDM scatter to out
  const float om = omega[d];
  float* so = lds + STORE_DW + wave * SBL;
  #pragma unroll
  for (int jj = 0; jj < 4; ++jj) {
    float2 p;
    const float z0 = acc[2 * jj    ] + om * Xd[2 * jj    ];
    const float z1 = acc[2 * jj + 1] + om * Xd[2 * jj + 1];
    p.x = z0 / (1.0f + expf(-z0));
    p.y = z1 / (1.0f + expf(-z1));
    *(float2*)(so + 16 * c + 8 * h + 2 * jj) = p;
  }
  asm volatile("s_wait_dscnt 0" ::: "memory");   // LDS results visible to TDM
  tdm_store_chan(out + (((size_t)t0) << 12) + baseChan,
                 (unsigned)((STORE_DW + wave * SBL) * 4));
  // S_ENDPGM's implicit wait-idle drains TENSORcnt before wave exit.
}

// ---------------------------------------------------------------- launch
extern "C" void kernel_launch(void* const* d_in, const int* in_sizes, int n_in,
                              void* d_out, int out_size, void* d_ws, size_t ws_size,
                              hipStream_t stream) {
  const float* x     = (const float*)d_in[0];
  const float* delta = (const float*)d_in[1];
  const float* alpha = (const float*)d_in[2];
  const float* beta  = (const float*)d_in[3];
  const float* gamma = (const float*)d_in[4];
  const float* omega = (const float*)d_in[5];
  float* out = (float*)d_out;
  float* ws  = (float*)d_ws;     // needs 12 MB (4 const + 4 F + 4 Sin)

  k_const<<<DD, 32, 0, stream>>>(delta, alpha, beta, gamma, ws);
  dim3 grid(NCH, 2);             // 8 waves/block, wave -> superblock
  k_pass1<<<grid, 256, LDS_LOAD_BYTES, stream>>>(x, ws);
  k_pass2<<<(NCH * NN) / 256, 256, 0, stream>>>(ws);
  k_pass3<<<grid, 256, LDS_ALL_BYTES, stream>>>(x, omega, ws, out);
}